// DKD_2594160246856
// MI455X (gfx1250) — compile-verified
//
#include <hip/hip_runtime.h>
#include <hip/hip_bf16.h>

#define B_    8
#define H_    2048
#define W_    2048
#define K_    8192
#define NBINS 2048
#define CAP   16384
#define TW    64
#define THT   8

typedef float v2f __attribute__((ext_vector_type(2)));
typedef float v8f __attribute__((ext_vector_type(8)));

// CDNA5 async global->LDS copy (ASYNCcnt-tracked), guarded so compile stays
// green on toolchains that lack the builtin. Probe confirmed the builtin's
// pointer params are int* in AS1 (global) / AS3 (LDS).
#if defined(__has_builtin)
#  if __has_builtin(__builtin_amdgcn_global_load_async_to_lds_b32)
#    define ASYNC_LDS 1
#  else
#    define ASYNC_LDS 0
#  endif
#  if __has_builtin(__builtin_amdgcn_s_wait_asynccnt)
#    define WAIT_ASYNC() __builtin_amdgcn_s_wait_asynccnt(0)
#  else
#    define WAIT_ASYNC() asm volatile("s_wait_asynccnt 0x0" ::: "memory")
#  endif
#else
#  define ASYNC_LDS 0
#  define WAIT_ASYNC()
#endif

typedef __attribute__((address_space(1))) int as1_int;
typedef __attribute__((address_space(3))) int as3_int;

__device__ __forceinline__ unsigned monoKey(float v) {
    unsigned fb = __float_as_uint(v);
    return fb ^ ((unsigned)((int)fb >> 31) | 0x80000000u);
}

// ---------------------------------------------------------------------------
// Pass 1/3: 5x5 NMS (separable max in LDS). COLLECT=0: per-batch value-bin
// histogram. COLLECT=1: compact candidates >= threshold bin as packed u64 keys.
// ---------------------------------------------------------------------------
template <int COLLECT>
__global__ void nms_pass(const float* __restrict__ scores,
                         unsigned* __restrict__ hist,
                         const unsigned* __restrict__ thr,
                         unsigned* __restrict__ cnt,
                         unsigned long long* __restrict__ cand)
{
    __shared__ float tile[THT + 4][TW + 4];
    __shared__ float hmax[THT + 4][TW];
    __shared__ unsigned lhist[NBINS];

    const int tid = threadIdx.x;
    const int b   = blockIdx.z;
    const int tx0 = blockIdx.x * TW;
    const int ty0 = blockIdx.y * THT;
    const float* img = scores + (size_t)b * ((size_t)H_ * W_);

    if (!COLLECT)
        for (int i = tid; i < NBINS; i += 256) lhist[i] = 0u;

    for (int i = tid; i < (THT + 4) * (TW + 4); i += 256) {
        int r = i / (TW + 4), c = i % (TW + 4);
        int gy = ty0 - 2 + r, gx = tx0 - 2 + c;
        if (gx >= 0 && gx < W_ && gy >= 0 && gy < H_) {
#if ASYNC_LDS
            __builtin_amdgcn_global_load_async_to_lds_b32(
                (as1_int*)(img + (size_t)gy * W_ + gx),
                (as3_int*)&tile[r][c],
                0, 0);
#else
            tile[r][c] = img[(size_t)gy * W_ + gx];
#endif
        } else {
            tile[r][c] = -__builtin_inff();   // halo outside image
        }
    }
    // CDNA5 global_prefetch_b8: warm next tile-row block in L2
    {
        int gy = ty0 + THT + (tid >> 5);
        int gx = tx0 + ((tid & 31) << 1);
        if (gy < H_) __builtin_prefetch(&img[(size_t)gy * W_ + gx], 0, 1);
    }
#if ASYNC_LDS
    WAIT_ASYNC();   // drain this wave's ASYNCcnt before the workgroup barrier
#endif
    __syncthreads();

    for (int i = tid; i < (THT + 4) * TW; i += 256) {
        int r = i / TW, c = i % TW;
        float m = tile[r][c];
        m = fmaxf(m, tile[r][c + 1]); m = fmaxf(m, tile[r][c + 2]);
        m = fmaxf(m, tile[r][c + 3]); m = fmaxf(m, tile[r][c + 4]);
        hmax[r][c] = m;
    }
    __syncthreads();

    unsigned thrKey = 0;
    if (COLLECT) thrKey = thr[b] << 21;

    for (int i = tid; i < THT * TW; i += 256) {
        int r = i / TW, c = i % TW;
        int gy = ty0 + r, gx = tx0 + c;
        if (gx < 2 || gx > W_ - 3 || gy < 2 || gy > H_ - 3) continue;  // border zeroed
        float v = tile[r + 2][c + 2];
        float m = hmax[r][c];
        m = fmaxf(m, hmax[r + 1][c]); m = fmaxf(m, hmax[r + 2][c]);
        m = fmaxf(m, hmax[r + 3][c]); m = fmaxf(m, hmax[r + 4][c]);
        if (v == m) {  // peak
            unsigned key = monoKey(v);
            if (!COLLECT) {
                atomicAdd(&lhist[key >> 21], 1u);
            } else if (key >= thrKey) {
                unsigned pos = atomicAdd(&cnt[b], 1u);
                if (pos < CAP) {
                    unsigned flat = (unsigned)(gy * W_ + gx);
                    // sort desc => value desc, then index asc (jax tie-break)
                    cand[(size_t)b * CAP + pos] =
                        ((unsigned long long)key << 32) | (unsigned)(~flat);
                }
            }
        }
    }

    if (!COLLECT) {
        __syncthreads();
        for (int i = tid; i < NBINS; i += 256)
            if (lhist[i]) atomicAdd(&hist[b * NBINS + i], lhist[i]);
    }
}

// ---------------------------------------------------------------------------
// Pass 2: find per-batch threshold bin t* (suffix count >= K)
// ---------------------------------------------------------------------------
__global__ void scan_hist(const unsigned* __restrict__ hist,
                          unsigned* __restrict__ thr)
{
    int b = threadIdx.x;
    if (b >= B_) return;
    unsigned cum = 0, t = 0;
    for (int i = NBINS - 1; i >= 0; --i) {
        cum += hist[b * NBINS + i];
        if (cum >= (unsigned)K_) { t = (unsigned)i; break; }
    }
    thr[b] = t;
}

// ---------------------------------------------------------------------------
// Pass 4: per-batch bitonic sort (desc) of CAP packed keys in LDS; emit top-K
// flat indices. 16384 * 8B = 128 KB LDS (CDNA5 WGP has 320 KB).
// ---------------------------------------------------------------------------
__global__ void __launch_bounds__(1024)
sort_emit(const unsigned* __restrict__ cnt,
          const unsigned long long* __restrict__ cand,
          int* __restrict__ kpIdx)
{
    __shared__ unsigned long long keys[CAP];
    const int tid = threadIdx.x;
    const int b = blockIdx.x;
    unsigned n = cnt[b]; if (n > CAP) n = CAP;
    for (int i = tid; i < CAP; i += 1024)
        keys[i] = (i < (int)n) ? cand[(size_t)b * CAP + i] : 0ull;
    __syncthreads();
    for (int k = 2; k <= CAP; k <<= 1) {
        for (int j = k >> 1; j > 0; j >>= 1) {
            for (int i = tid; i < CAP; i += 1024) {
                int ixj = i ^ j;
                if (ixj > i) {
                    unsigned long long a = keys[i], bb = keys[ixj];
                    bool asc = (i & k) != 0;
                    bool sw  = asc ? (a > bb) : (a < bb);
                    if (sw) { keys[i] = bb; keys[ixj] = a; }
                }
            }
            __syncthreads();
        }
    }
    for (int i = tid; i < K_; i += 1024) {
        unsigned long long kk = keys[i];
        unsigned flat = ~(unsigned)kk;
        if (kk == 0ull) flat = 2u * W_ + 2u;   // degenerate pad: safe interior px
        kpIdx[b * K_ + i] = (int)flat;
    }
}

// ---------------------------------------------------------------------------
// Pass 5: soft-argmax refinement via V_WMMA_F32_16X16X4_F32.
// One wave handles 16 keypoints: A = exp-weights (16 x 28, K-tiled by 4),
// B cols = [1, dx, dy, dx^2+dy^2], C row m = (S0, Sx, Sy, Sq) for keypoint m.
// ---------------------------------------------------------------------------
// Straight-line (branch-free) B-matrix generator: compiles to v_cndmask chain.
__device__ __forceinline__ float gval(int s, int n) {
    int   q  = s / 5;
    float dx = (float)(s - 5 * q - 2);
    float dy = (float)(q - 2);
    float g  = (n == 3) ? (dx * dx + dy * dy) : 0.0f;
    g = (n == 2) ? dy : g;
    g = (n == 1) ? dx : g;
    g = (n == 0) ? 1.0f : g;
    g = (s < 25) ? g : 0.0f;
    return g;
}

__global__ void __launch_bounds__(256)
refine(const float* __restrict__ scores,
       const int* __restrict__ kpIdx,
       float* __restrict__ out_kxy,
       float* __restrict__ out_score,
       float* __restrict__ out_disp)
{
    __shared__ float stats[8][16][4];
    const int tid  = threadIdx.x;
    const int lane = tid & 31;
    const int wv   = tid >> 5;
    const int m    = lane & 15;   // keypoint row M
    const int h    = lane >> 4;   // half-wave selects K slots {2h,2h+1} per chunk
    const int g    = (blockIdx.x * 8 + wv) * 16 + m;   // global keypoint id
    const int b    = g >> 13;                          // / K_
    const float* img = scores + (size_t)b * ((size_t)H_ * W_);

    int idx = kpIdx[g];
    int ky = idx / W_, kx = idx % W_;
    ky = min(max(ky, 2), H_ - 3);
    kx = min(max(kx, 2), W_ - 3);   // interior => 5x5 patch always in-bounds

    // gather this lane's 14 patch slots (slot s = 4j + 2h + t, s<25 real)
    float val[14];
    float lmax = -__builtin_inff();
    #pragma unroll
    for (int j = 0; j < 7; ++j) {
        #pragma unroll
        for (int t = 0; t < 2; ++t) {
            int s = 4 * j + 2 * h + t;
            float v = -__builtin_inff();
            if (s < 25) {
                int di = s / 5 - 2, dj = s % 5 - 2;
                v = img[(size_t)(ky + di) * W_ + (kx + dj)];
                lmax = fmaxf(lmax, v);
            }
            val[2 * j + t] = v;
        }
    }
    // two half-waves jointly cover all 25 slots of row m
    float maxv = fmaxf(lmax, __shfl_xor(lmax, 16, 32));

    const float SCALE = 14.426950408889634f;  // log2(e) / T, T = 0.1
    v8f acc = {0.f, 0.f, 0.f, 0.f, 0.f, 0.f, 0.f, 0.f};
    #pragma unroll
    for (int j = 0; j < 7; ++j) {
        int s0 = 4 * j + 2 * h, s1 = s0 + 1;
        v2f a, bv;
        a.x = (s0 < 25) ? exp2f((val[2 * j]     - maxv) * SCALE) : 0.0f;
        a.y = (s1 < 25) ? exp2f((val[2 * j + 1] - maxv) * SCALE) : 0.0f;
        bv.x = gval(s0, m);   // B row s0, column N = lane&15
        bv.y = gval(s1, m);
        acc = __builtin_amdgcn_wmma_f32_16x16x4_f32(
            false, a, false, bv, (short)0, acc, false, false);
    }

    // C layout: lane holds column N=m for rows M = 8h + r; stash cols 0..3
    if (m < 4) {
        #pragma unroll
        for (int r = 0; r < 8; ++r) stats[wv][8 * h + r][m] = acc[r];
    }
    __syncthreads();

    if (lane < 16) {   // lane == its keypoint m; kx/ky already local
        float S0 = stats[wv][lane][0];
        float Sx = stats[wv][lane][1];
        float Sy = stats[wv][lane][2];
        float Sq = stats[wv][lane][3];
        float inv = 1.0f / S0;
        float xr = Sx * inv, yr = Sy * inv;
        // disp = sum e*(|g-xy|^2)/r^2 / sum e = (Sq/S0 - |xy|^2) / r^2
        float disp = (Sq * inv - (xr * xr + yr * yr)) * 0.25f;

        float fx = (float)kx + xr, fy = (float)ky + yr;
        float kxx = fx * (2.0f / (float)(W_ - 1)) - 1.0f;
        float kyy = fy * (2.0f / (float)(H_ - 1)) - 1.0f;

        // bilinear resample, align_corners=True
        float px = (kxx + 1.0f) * 0.5f * (float)(W_ - 1);
        float py = (kyy + 1.0f) * 0.5f * (float)(H_ - 1);
        int x0 = (int)floorf(px); x0 = min(max(x0, 0), W_ - 1);
        int y0 = (int)floorf(py); y0 = min(max(y0, 0), H_ - 1);
        int x1 = min(x0 + 1, W_ - 1);
        int y1 = min(y0 + 1, H_ - 1);
        float wx = px - (float)x0, wy = py - (float)y0;
        float v00 = img[(size_t)y0 * W_ + x0], v01 = img[(size_t)y0 * W_ + x1];
        float v10 = img[(size_t)y1 * W_ + x0], v11 = img[(size_t)y1 * W_ + x1];
        float sc = v00 * (1.f - wx) * (1.f - wy) + v01 * wx * (1.f - wy)
                 + v10 * (1.f - wx) * wy + v11 * wx * wy;

        out_kxy[(size_t)g * 2 + 0] = kxx;
        out_kxy[(size_t)g * 2 + 1] = kyy;
        out_score[g] = sc;
        out_disp[g]  = disp;
    }
}

// ---------------------------------------------------------------------------
extern "C" void kernel_launch(void* const* d_in, const int* in_sizes, int n_in,
                              void* d_out, int out_size, void* d_ws, size_t ws_size,
                              hipStream_t stream)
{
    const float* scores = (const float*)d_in[0];

    char* ws = (char*)d_ws;
    unsigned* hist           = (unsigned*)(ws);                 // 8*2048*4 = 65536 B
    unsigned* thr            = (unsigned*)(ws + 65536);         // 32 B
    unsigned* cnt            = (unsigned*)(ws + 65568);         // 32 B
    unsigned long long* cand = (unsigned long long*)(ws + 65600);        // 8*16384*8 = 1 MiB
    int* kpIdx               = (int*)(ws + 65600 + (size_t)B_ * CAP * 8); // 8*8192*4

    float* out_kxy = (float*)d_out;                     // (8,8192,2)
    float* out_sc  = out_kxy + (size_t)B_ * K_ * 2;     // (8,8192)
    float* out_dp  = out_sc  + (size_t)B_ * K_;         // (8,8192)

    (void)hipMemsetAsync(d_ws, 0, 65600, stream);       // hist + thr + cnt

    dim3 gridT(W_ / TW, H_ / THT, B_);
    nms_pass<0><<<gridT, 256, 0, stream>>>(scores, hist, thr, cnt, cand);
    scan_hist  <<<1, 32, 0, stream>>>(hist, thr);
    nms_pass<1><<<gridT, 256, 0, stream>>>(scores, hist, thr, cnt, cand);
    sort_emit  <<<B_, 1024, 0, stream>>>(cnt, cand, kpIdx);
    refine     <<<(B_ * K_) / (8 * 16), 256, 0, stream>>>(scores, kpIdx,
                                                          out_kxy, out_sc, out_dp);
}